// GMM_64965675319738
// MI455X (gfx1250) — compile-verified
//
#include <hip/hip_runtime.h>
#include <math.h>

typedef __attribute__((ext_vector_type(2))) float v2f;
typedef __attribute__((ext_vector_type(8))) float v8f;

#define KCOMP 64
#define LOG2PI 1.8378770664093453f
#define ROWSTRIDE 67   // 32 rows per wave, stride 67: conflict-free tile store AND row read

// ---------------------------------------------------------------------------
// Prep: fold tril -> cov -> cholesky -> inverse -> precision matrix and
// log_softmax(pi) into an 8x64 f32 weight matrix W (feature-major rows):
//   f(x) = [x0^2, x0*x1, x1^2, x0, x1, 1, 0, 0]
//   weighted[n,k] = f(x_n) . W[:,k]
// ---------------------------------------------------------------------------
__global__ __launch_bounds__(KCOMP) void gmm_prep_kernel(
    const float* __restrict__ means,     // [K,2]
    const float* __restrict__ chol_var,  // [K,2,2]
    const float* __restrict__ pi,        // [K]
    float* __restrict__ W)               // [8,K]
{
    int k = threadIdx.x;

    // log_softmax normalizer (K=64, redundantly per thread — trivial)
    float m = -INFINITY;
    for (int i = 0; i < KCOMP; ++i) m = fmaxf(m, pi[i]);
    float s = 0.f;
    for (int i = 0; i < KCOMP; ++i) s += __expf(pi[i] - m);
    float lse = m + __logf(s);

    // L = tril(chol_var[k])
    float l00 = chol_var[k*4 + 0];
    float l10 = chol_var[k*4 + 2];
    float l11 = chol_var[k*4 + 3];
    const float EPSV = 1e-6f;

    // cov = L L^T + eps I (2x2 symmetric)
    float c00 = l00*l00 + EPSV;
    float c10 = l10*l00;
    float c11 = l10*l10 + l11*l11 + EPSV;

    // cholesky of cov
    float s00 = sqrtf(c00);
    float s10 = c10 / s00;
    float s11 = sqrtf(c11 - s10*s10);
    float logdet = 2.f * (__logf(s00) + __logf(s11));

    // inverse of lower-tri chol
    float i00 = 1.f / s00;
    float i11 = 1.f / s11;
    float i10 = -s10 * i00 * i11;

    // precision A = cinv^T cinv
    float a00 = i00*i00 + i10*i10;
    float a01 = i10*i11;
    float a11 = i11*i11;

    float mu0 = means[k*2 + 0], mu1 = means[k*2 + 1];
    float b0 = a00*mu0 + a01*mu1;
    float b1 = a01*mu0 + a11*mu1;
    float cterm = mu0*b0 + mu1*b1;

    W[0*KCOMP + k] = -0.5f * a00;
    W[1*KCOMP + k] = -a01;
    W[2*KCOMP + k] = -0.5f * a11;
    W[3*KCOMP + k] = b0;
    W[4*KCOMP + k] = b1;
    W[5*KCOMP + k] = -0.5f * (2.f*LOG2PI + logdet + cterm) + (pi[k] - lse);
    W[6*KCOMP + k] = 0.f;
    W[7*KCOMP + k] = 0.f;
}

// ---------------------------------------------------------------------------
// Main: [N x 8] . [8 x 64] via V_WMMA_F32_16X16X4_F32 (two K=4 steps),
// then per-row logsumexp over the 64 components.
// Each wave owns 32 points (two 16-row tiles); 4 waves / block = 128 points.
// ---------------------------------------------------------------------------
__global__ __launch_bounds__(128) void gmm_main_kernel(
    const float* __restrict__ x,   // [N,2]
    const float* __restrict__ Wg,  // [8,64]
    float* __restrict__ out,       // [N]
    int N)
{
    __shared__ float Wl[8 * KCOMP];
    __shared__ float buf[4][32 * ROWSTRIDE];

    int tid = threadIdx.x;
    for (int i = tid; i < 8 * KCOMP; i += 128) Wl[i] = Wg[i];
    __syncthreads();

    int wave = tid >> 5;
    int lane = tid & 31;
    int l15  = lane & 15;
    bool hi  = lane >= 16;
    int base = blockIdx.x * 128 + wave * 32;

    // B fragments for all 4 column tiles.
    // f32 B 4x16 layout: lane<16 holds K=0,1 in vgpr0/1 for col=lane;
    // lane>=16 holds K=2,3 for col=lane-16.
    v2f b0[4], b1[4];
#pragma unroll
    for (int t = 0; t < 4; ++t) {
        int c = t * 16 + l15;
        b0[t].x = Wl[(hi ? 2 : 0) * KCOMP + c];
        b0[t].y = Wl[(hi ? 3 : 1) * KCOMP + c];
        b1[t].x = Wl[(hi ? 6 : 4) * KCOMP + c];
        b1[t].y = Wl[(hi ? 7 : 5) * KCOMP + c];
    }

    const float2* x2 = (const float2*)x;

#pragma unroll
    for (int rt = 0; rt < 2; ++rt) {
        // A fragment: row M = l15 of this 16-row tile.
        // lane<16: features {0,1}; lane>=16: features {2,3} (and {4,5}/{6,7}).
        int p = base + rt * 16 + l15;
        p = p < N ? p : N - 1;          // clamp: keep EXEC all-ones for WMMA
        float2 xv = x2[p];

        v2f a0, a1;
        a0.x = hi ? xv.y * xv.y : xv.x * xv.x;   // f2 : f0
        a0.y = hi ? xv.x        : xv.x * xv.y;   // f3 : f1
        a1.x = hi ? 0.f         : xv.y;          // f6 : f4
        a1.y = hi ? 0.f         : 1.f;           // f7 : f5

#pragma unroll
        for (int t = 0; t < 4; ++t) {
            v8f acc = {};
            acc = __builtin_amdgcn_wmma_f32_16x16x4_f32(
                false, a0, false, b0[t], (short)0, acc, false, false);
            acc = __builtin_amdgcn_wmma_f32_16x16x4_f32(
                false, a1, false, b1[t], (short)0, acc, false, false);

            // D layout: vgpr r, lane<16 -> (row r, col lane); lane>=16 -> (row r+8)
            int col   = t * 16 + l15;
            int rbase = rt * 16 + (hi ? 8 : 0);
#pragma unroll
            for (int r = 0; r < 8; ++r)
                buf[wave][(rbase + r) * ROWSTRIDE + col] = acc[r];
        }
    }
    __syncthreads();

    // Each lane reduces its own row of 64 weighted log-probs.
    const float* rowp = &buf[wave][lane * ROWSTRIDE];
    float mx = -INFINITY;
#pragma unroll 8
    for (int j = 0; j < KCOMP; ++j) mx = fmaxf(mx, rowp[j]);
    float ssum = 0.f;
#pragma unroll 8
    for (int j = 0; j < KCOMP; ++j) ssum += __expf(rowp[j] - mx);

    int pout = base + lane;
    if (pout < N) out[pout] = mx + __logf(ssum);
}

extern "C" void kernel_launch(void* const* d_in, const int* in_sizes, int n_in,
                              void* d_out, int out_size, void* d_ws, size_t ws_size,
                              hipStream_t stream) {
    const float* x        = (const float*)d_in[0];  // [N,2]
    const float* means    = (const float*)d_in[1];  // [64,2]
    const float* chol_var = (const float*)d_in[2];  // [64,2,2]
    const float* pi       = (const float*)d_in[3];  // [64]
    float* W   = (float*)d_ws;                      // 8*64 floats scratch
    float* out = (float*)d_out;

    int N = in_sizes[0] / 2;

    gmm_prep_kernel<<<1, KCOMP, 0, stream>>>(means, chol_var, pi, W);

    int blocks = (N + 127) / 128;
    gmm_main_kernel<<<blocks, 128, 0, stream>>>(x, W, out, N);
}